// DGCNN_21225728377068
// MI455X (gfx1250) — compile-verified
//
#include <hip/hip_runtime.h>

typedef __attribute__((ext_vector_type(2))) float v2f;
typedef __attribute__((ext_vector_type(8))) float v8f;

#define DEVINL __device__ __forceinline__

constexpr int Bn = 8;
constexpr int Np = 4096;
constexpr int Kn = 20;
constexpr float NEG_INF = -3.402823466e38f;
constexpr float POS_INF = 3.402823466e38f;

DEVINL v2f ld2(const float* __restrict__ p) { return *(const v2f*)p; }

// ---------------------------------------------------------------------------
// KNN: one thread per query point; whole batch's coords staged in LDS (48KB).
// Register-resident, fully unrolled top-20 insertion list (matches top_k of
// -dist semantics: smallest squared distance, ties -> lower index first).
// Also emits transposed points p_t[b][n][3] for the first EdgeConv.
// ---------------------------------------------------------------------------
__global__ __launch_bounds__(256) void knn_kernel(const float* __restrict__ pts,
                                                  int* __restrict__ knn,
                                                  float* __restrict__ p_t) {
  __shared__ float sx[Np], sy[Np], sz[Np];
  const int b = blockIdx.x / (Np / 256);
  const int n = (blockIdx.x % (Np / 256)) * 256 + threadIdx.x;
  const float* px = pts + (size_t)b * 3 * Np;
  for (int j = threadIdx.x; j < Np; j += 256) {
    sx[j] = px[j];
    sy[j] = px[Np + j];
    sz[j] = px[2 * Np + j];
  }
  __syncthreads();

  const float qx = sx[n], qy = sy[n], qz = sz[n];
  float dl[Kn];
  int il[Kn];
#pragma unroll
  for (int t = 0; t < Kn; ++t) { dl[t] = POS_INF; il[t] = 0; }

  for (int j = 0; j < Np; ++j) {
    const float dx = qx - sx[j], dy = qy - sy[j], dz = qz - sz[j];
    const float dist = dx * dx + dy * dy + dz * dz;
    if (dist < dl[Kn - 1]) {
      float cd = dist; int ci = j;
#pragma unroll
      for (int p = 0; p < Kn; ++p) {
        if (cd < dl[p]) {
          const float td = dl[p]; const int ti = il[p];
          dl[p] = cd; il[p] = ci;
          cd = td; ci = ti;
        }
      }
    }
  }

  const size_t base = (size_t)(b * Np + n) * Kn;
#pragma unroll
  for (int t = 0; t < Kn; ++t) knn[base + t] = il[t];

  float* pr = p_t + (size_t)(b * Np + n) * 3;
  pr[0] = qx; pr[1] = qy; pr[2] = qz;
}

// Generic edge element (only used by the CIN=3 layer, K padded 6 -> 8).
template <int CIN>
DEVINL float edge_val(const float* __restrict__ ctr, const float* __restrict__ nbr, int k) {
  if (k < CIN) return ctr[k];
  const int c = k - CIN;
  if (c < CIN) return nbr[c] - ctr[c];
  return 0.0f;
}

// ---------------------------------------------------------------------------
// EdgeConv via V_WMMA_F32_16X16X4_F32.
// One wave per point. Columns = 20 neighbor slots padded to 32 = two 16-col
// WMMA tiles. A tile = 16 rows of W, K-dim = 2*CIN stepped by 4.
// Per ISA layouts (wave32):
//   A: lane L -> m = L&15, a.x = W[m][k0+2h], a.y = W[m][k0+2h+1], h = L>=16
//   B: lane L -> col = L&15, b.x = E[k0+2h][col], b.y = E[k0+2h+1][col]
//   D: reg r at lane L -> row r + 8h, col L&15
// For CIN%4==0 the edge matrix is processed as one fused loop issuing two
// WMMAs per 4-wide K step: W[:,k].ctr  and  W[:,CIN+k].(nbr-ctr), with pure
// b64 operand loads (ctr chunk reused for the subtraction).
// Epilogue: relu(s*y+b), mask pad slots to -inf, butterfly max over 16 cols
// within each half-wave, lane 0 / lane 16 write the 16 output channels.
// ---------------------------------------------------------------------------
template <int CIN, int COUT, int INS, int OUTS>
__global__ __launch_bounds__(256) void edgeconv_kernel(const float* __restrict__ in,
                                                       const int* __restrict__ knn,
                                                       const float* __restrict__ W,
                                                       const float* __restrict__ sc,
                                                       const float* __restrict__ bi,
                                                       float* __restrict__ out) {
  constexpr int KD = 2 * CIN;
  constexpr int KTOT = (KD + 3) & ~3;

  const int wid = blockIdx.x * (blockDim.x >> 5) + (threadIdx.x >> 5);
  const int b = wid >> 12;          // / Np
  const int n = wid & (Np - 1);
  const int lane = threadIdx.x & 31;
  const int half = lane >> 4;
  const int l15 = lane & 15;

  const float* ctr = in + (size_t)(b * Np + n) * INS;
  const int* kn = knn + (size_t)(b * Np + n) * Kn;

#pragma unroll 1
  for (int mt = 0; mt < COUT / 16; ++mt) {
    float pm[8];
#pragma unroll
    for (int r = 0; r < 8; ++r) pm[r] = NEG_INF;

#pragma unroll 1
    for (int tile = 0; tile < 2; ++tile) {
      const int s = tile * 16 + l15;
      const bool valid = (s < Kn);
      const int j = valid ? kn[s] : n;
      const float* nbr = in + (size_t)(b * Np + j) * INS;
      __builtin_prefetch(nbr, 0, 1);

      v8f acc = {0.f, 0.f, 0.f, 0.f, 0.f, 0.f, 0.f, 0.f};
      const float* wrow = W + (size_t)(mt * 16 + l15) * KD;

      if constexpr ((CIN & 3) == 0) {
        // Fused: per K-chunk issue ctr-region and nbr-region WMMAs together.
#pragma unroll
        for (int k0 = 0; k0 < CIN; k0 += 4) {
          const int kk = k0 + 2 * half;
          const v2f c2 = ld2(ctr + kk);        // uniform row, cacheline hit
          const v2f a0 = ld2(wrow + kk);
          acc = __builtin_amdgcn_wmma_f32_16x16x4_f32(false, a0, false, c2,
                                                      (short)0, acc, false, false);
          const v2f n2 = ld2(nbr + kk);        // the K-NN gather
          const v2f a1 = ld2(wrow + CIN + kk);
          const v2f d2 = n2 - c2;
          acc = __builtin_amdgcn_wmma_f32_16x16x4_f32(false, a1, false, d2,
                                                      (short)0, acc, false, false);
        }
      } else {
        // Generic padded path (CIN == 3, KTOT == 8).
#pragma unroll
        for (int k0 = 0; k0 < KTOT; k0 += 4) {
          const int kk = k0 + 2 * half;
          v2f a, bb;
          a.x = (kk < KD) ? wrow[kk] : 0.0f;
          a.y = (kk + 1 < KD) ? wrow[kk + 1] : 0.0f;
          bb.x = edge_val<CIN>(ctr, nbr, kk);
          bb.y = edge_val<CIN>(ctr, nbr, kk + 1);
          acc = __builtin_amdgcn_wmma_f32_16x16x4_f32(false, a, false, bb,
                                                      (short)0, acc, false, false);
        }
      }

#pragma unroll
      for (int r = 0; r < 8; ++r) {
        const int m = mt * 16 + 8 * half + r;
        float y = fmaxf(sc[m] * acc[r] + bi[m], 0.0f);
        y = valid ? y : NEG_INF;
        pm[r] = fmaxf(pm[r], y);
      }
    }

#pragma unroll
    for (int r = 0; r < 8; ++r) {
#pragma unroll
      for (int off = 1; off < 16; off <<= 1)
        pm[r] = fmaxf(pm[r], __shfl_xor(pm[r], off, 32));
    }

    if (l15 == 0) {
      float* orow = out + (size_t)(b * Np + n) * OUTS + mt * 16 + 8 * half;
#pragma unroll
      for (int r = 0; r < 8; ++r) orow[r] = pm[r];
    }
  }
}

// ---------------------------------------------------------------------------
// Local layer: 1024 x 320 GEMM over all points, WMMA-tiled, fused partial
// max/argmax over each 16-point column tile (branchless value+index
// butterfly -> v_cmp/v_cndmask, no EXEC divergence).
// Grid: B * (N/16) blocks of 256 threads; wave handles m-tiles strided by 8.
// ---------------------------------------------------------------------------
__global__ __launch_bounds__(256) void local_kernel(const float* __restrict__ cat,
                                                    const float* __restrict__ W,
                                                    const float* __restrict__ sc,
                                                    const float* __restrict__ bi,
                                                    float* __restrict__ pmax,
                                                    int* __restrict__ pidx) {
  constexpr int CO = 1024, KD = 320, NT = Np / 16;
  const int ct = blockIdx.x & (NT - 1);
  const int b = blockIdx.x >> 8;
  const int wave = threadIdx.x >> 5;
  const int lane = threadIdx.x & 31;
  const int half = lane >> 4;
  const int l15 = lane & 15;
  const int n = ct * 16 + l15;
  const float* col = cat + (size_t)(b * Np + n) * KD;

#pragma unroll 1
  for (int mt = wave; mt < CO / 16; mt += 8) {
    v8f acc = {0.f, 0.f, 0.f, 0.f, 0.f, 0.f, 0.f, 0.f};
    const float* wrow = W + (size_t)(mt * 16 + l15) * KD;
#pragma unroll 4
    for (int k0 = 0; k0 < KD; k0 += 4) {
      const int kk = k0 + 2 * half;
      const v2f a = ld2(wrow + kk);
      const v2f bb = ld2(col + kk);
      acc = __builtin_amdgcn_wmma_f32_16x16x4_f32(false, a, false, bb,
                                                  (short)0, acc, false, false);
    }

#pragma unroll
    for (int r = 0; r < 8; ++r) {
      const int m = mt * 16 + 8 * half + r;
      float y = fmaxf(sc[m] * acc[r] + bi[m], 0.0f);
      int yi = n;
#pragma unroll
      for (int off = 1; off < 16; off <<= 1) {
        const float ov = __shfl_xor(y, off, 32);
        const int oi = __shfl_xor(yi, off, 32);
        const bool take = (ov > y) | ((ov == y) & (oi < yi));
        y = take ? ov : y;
        yi = take ? oi : yi;
      }
      if (l15 == 0) {
        pmax[(size_t)(b * CO + m) * NT + ct] = y;
        pidx[(size_t)(b * CO + m) * NT + ct] = yi;
      }
    }
  }
}

// Final max/argmax across the 256 column-tile partials per (b, channel).
__global__ void reduce_kernel(const float* __restrict__ pmax,
                              const int* __restrict__ pidx,
                              float* __restrict__ glob,
                              float* __restrict__ out_idx) {
  const int t = blockIdx.x * blockDim.x + threadIdx.x;
  if (t >= Bn * 1024) return;
  float best = NEG_INF; int bix = 0;
  for (int c = 0; c < Np / 16; ++c) {
    const float v = pmax[(size_t)t * (Np / 16) + c];
    const int iv = pidx[(size_t)t * (Np / 16) + c];
    const bool take = (v > best) | ((v == best) & (iv < bix));
    best = take ? v : best;
    bix = take ? iv : bix;
  }
  glob[t] = best;
  out_idx[t] = (float)bix;
}

__global__ void mlp1_kernel(const float* __restrict__ glob, const float* __restrict__ W,
                            const float* __restrict__ sc, const float* __restrict__ bi,
                            float* __restrict__ h) {
  const int t = blockIdx.x * blockDim.x + threadIdx.x;
  if (t >= Bn * 512) return;
  const int b = t >> 9, o = t & 511;
  const float* g = glob + (size_t)b * 1024;
  const float* w = W + (size_t)o * 1024;
  float acc = 0.0f;
  for (int c = 0; c < 1024; ++c) acc += g[c] * w[c];
  h[t] = fmaxf(sc[o] * acc + bi[o], 0.0f);
}

__global__ void mlp2_kernel(const float* __restrict__ h, const float* __restrict__ W,
                            const float* __restrict__ sc, const float* __restrict__ bi,
                            float* __restrict__ out) {
  const int t = blockIdx.x * blockDim.x + threadIdx.x;
  if (t >= Bn * 256) return;
  const int b = t >> 8, o = t & 255;
  const float* hh = h + (size_t)b * 512;
  const float* w = W + (size_t)o * 512;
  float acc = 0.0f;
  for (int c = 0; c < 512; ++c) acc += hh[c] * w[c];
  out[t] = fmaxf(sc[o] * acc + bi[o], 0.0f);
}

// ---------------------------------------------------------------------------
extern "C" void kernel_launch(void* const* d_in, const int* in_sizes, int n_in,
                              void* d_out, int out_size, void* d_ws, size_t ws_size,
                              hipStream_t stream) {
  (void)in_sizes; (void)n_in; (void)out_size; (void)ws_size;

  const float* points = (const float*)d_in[0];
  const float* ec_w[4] = {(const float*)d_in[1], (const float*)d_in[4],
                          (const float*)d_in[7], (const float*)d_in[10]};
  const float* ec_s[4] = {(const float*)d_in[2], (const float*)d_in[5],
                          (const float*)d_in[8], (const float*)d_in[11]};
  const float* ec_b[4] = {(const float*)d_in[3], (const float*)d_in[6],
                          (const float*)d_in[9], (const float*)d_in[12]};
  const float* local_w = (const float*)d_in[13];
  const float* local_s = (const float*)d_in[14];
  const float* local_b = (const float*)d_in[15];
  const float* g_w0 = (const float*)d_in[16];
  const float* g_s0 = (const float*)d_in[17];
  const float* g_b0 = (const float*)d_in[18];
  const float* g_w1 = (const float*)d_in[19];
  const float* g_s1 = (const float*)d_in[20];
  const float* g_b1 = (const float*)d_in[21];

  float* out = (float*)d_out;

  // Workspace carve-up (256B aligned).
  char* ws = (char*)d_ws;
  size_t off = 0;
  auto carve = [&](size_t bytes) -> char* {
    char* p = ws + off;
    off = (off + bytes + 255) & ~(size_t)255;
    return p;
  };
  int*   knn  = (int*)carve((size_t)Bn * Np * Kn * 4);
  float* p_t  = (float*)carve((size_t)Bn * Np * 3 * 4);
  float* cat  = (float*)carve((size_t)Bn * Np * 320 * 4);
  float* pmax = (float*)carve((size_t)Bn * 1024 * (Np / 16) * 4);
  int*   pidx = (int*)carve((size_t)Bn * 1024 * (Np / 16) * 4);
  float* glob = (float*)carve((size_t)Bn * 1024 * 4);
  float* hbuf = (float*)carve((size_t)Bn * 512 * 4);

  // 1) KNN + point transpose
  knn_kernel<<<Bn * (Np / 256), 256, 0, stream>>>(points, knn, p_t);

  // 2) EdgeConv stack; outputs written straight into concat slices
  const int ecgrid = Bn * Np / 8;  // one wave per point, 8 waves/block
  edgeconv_kernel<3, 64, 3, 320><<<ecgrid, 256, 0, stream>>>(
      p_t, knn, ec_w[0], ec_s[0], ec_b[0], cat + 0);
  edgeconv_kernel<64, 64, 320, 320><<<ecgrid, 256, 0, stream>>>(
      cat + 0, knn, ec_w[1], ec_s[1], ec_b[1], cat + 64);
  edgeconv_kernel<64, 64, 320, 320><<<ecgrid, 256, 0, stream>>>(
      cat + 64, knn, ec_w[2], ec_s[2], ec_b[2], cat + 128);
  edgeconv_kernel<64, 128, 320, 320><<<ecgrid, 256, 0, stream>>>(
      cat + 128, knn, ec_w[3], ec_s[3], ec_b[3], cat + 192);

  // 3) Local 1024x320 layer with fused per-tile max/argmax
  local_kernel<<<Bn * (Np / 16), 256, 0, stream>>>(cat, local_w, local_s, local_b,
                                                   pmax, pidx);
  reduce_kernel<<<(Bn * 1024 + 255) / 256, 256, 0, stream>>>(pmax, pidx, glob,
                                                             out + Bn * 256);

  // 4) Global MLP head
  mlp1_kernel<<<(Bn * 512 + 255) / 256, 256, 0, stream>>>(glob, g_w0, g_s0, g_b0, hbuf);
  mlp2_kernel<<<(Bn * 256 + 255) / 256, 256, 0, stream>>>(hbuf, g_w1, g_s1, g_b1, out);
}